// S_MUGCN_51032801411522
// MI455X (gfx1250) — compile-verified
//
#include <hip/hip_runtime.h>

#define DIMS 128
#define LDS_STRIDE 132   // padded stride for transposed W in LDS (bank-conflict avoidance)

typedef __attribute__((ext_vector_type(2))) float v2f;
typedef __attribute__((ext_vector_type(8))) float v8f;

// ---------------------------------------------------------------------------
// H[n x 128] = X[n x 128] @ W[128 x 128]   (fp32 WMMA 16x16x4)
// Block: 256 threads = 8 waves; each wave computes a 16-row strip (block: 128 rows).
// W is staged transposed into LDS so the B operand is a contiguous ds_load_b64.
// ---------------------------------------------------------------------------
__global__ __launch_bounds__(256)
void gemm_wmma_f32(const float* __restrict__ X, const float* __restrict__ W,
                   float* __restrict__ H, int n) {
    __shared__ float ldsWT[DIMS * LDS_STRIDE];

    // cooperative transpose-load: ldsWT[col][k] = W[k][col]
    for (int idx = threadIdx.x; idx < DIMS * DIMS; idx += 256) {
        int k = idx >> 7;
        int c = idx & 127;
        ldsWT[c * LDS_STRIDE + k] = W[idx];
    }
    __syncthreads();

    const int wave = threadIdx.x >> 5;
    const int lane = threadIdx.x & 31;
    const int m0 = blockIdx.x * 128 + wave * 16;
    if (m0 >= n) return;                 // wave-uniform: EXEC stays all-1s for WMMA

    const int lr = lane & 15;            // sub-row / column lane id
    const int kh = (lane >> 4) * 2;      // K sub-offset: 0 (lanes 0-15) or 2 (lanes 16-31)

    int row = m0 + lr;
    if (row >= n) row = n - 1;           // clamp loads; stores are guarded below

    v8f acc[8] = {};                     // 8 N-tiles of 16 -> full 128 cols

    const float* xrow = X + (size_t)row * DIMS;
    for (int k0 = 0; k0 < DIMS; k0 += 4) {
        // A 16x4 fp32: lane holds X[row][k0+kh .. k0+kh+1]
        v2f a = *(const v2f*)(xrow + k0 + kh);
#pragma unroll
        for (int t = 0; t < 8; ++t) {
            const int ncol = t * 16 + lr;
            // B 4x16 fp32: lane holds W[k0+kh .. k0+kh+1][ncol] (transposed in LDS)
            v2f b = *(const v2f*)(&ldsWT[ncol * LDS_STRIDE + k0 + kh]);
            acc[t] = __builtin_amdgcn_wmma_f32_16x16x4_f32(
                false, a, false, b, (short)0, acc[t], false, false);
        }
    }

    // C layout: VGPR v -> M = v (lanes 0-15) or v+8 (lanes 16-31); N = t*16 + lr
    const int mBase = m0 + (lane >> 4) * 8;
#pragma unroll
    for (int v = 0; v < 8; ++v) {
        const int m = mBase + v;
        if (m < n) {
            float* hrow = H + (size_t)m * DIMS;
#pragma unroll
            for (int t = 0; t < 8; ++t) hrow[t * 16 + lr] = acc[t][v];
        }
    }
}

// ---------------------------------------------------------------------------
// Degree / normalization helpers
// ---------------------------------------------------------------------------
__global__ void k_deg_init(float* __restrict__ deg, int n) {
    int i = blockIdx.x * blockDim.x + threadIdx.x;
    if (i < n) deg[i] = 1.0f;            // self-loop contribution
}

__global__ void k_deg_count(const int* __restrict__ dst, float* __restrict__ deg, int e) {
    int i = blockIdx.x * blockDim.x + threadIdx.x;
    if (i < e) atomicAdd(&deg[dst[i]], 1.0f);
}

__global__ void k_dinv(const float* __restrict__ deg, float* __restrict__ dinv, int n) {
    int i = blockIdx.x * blockDim.x + threadIdx.x;
    if (i < n) dinv[i] = rsqrtf(deg[i]);
}

// ---------------------------------------------------------------------------
// out[i] = b + dinv[i]^2 * H[i]   (self-loop term + bias; also initializes out)
// one thread per (node, 4 features)
// ---------------------------------------------------------------------------
__global__ void k_self_bias(const float* __restrict__ H, const float* __restrict__ dinv,
                            const float* __restrict__ b, float* __restrict__ out, int n) {
    int idx = blockIdx.x * blockDim.x + threadIdx.x;
    int i = idx >> 5;
    if (i >= n) return;
    int c0 = (idx & 31) * 4;
    float di = dinv[i];
    float w = di * di;
    const float4 h  = *(const float4*)(H + (size_t)i * DIMS + c0);
    const float4 bb = *(const float4*)(b + c0);
    float4 o;
    o.x = bb.x + h.x * w;
    o.y = bb.y + h.y * w;
    o.z = bb.z + h.z * w;
    o.w = bb.w + h.w * w;
    *(float4*)(out + (size_t)i * DIMS + c0) = o;
}

// ---------------------------------------------------------------------------
// Edge scatter: out[dst] += dinv[src]*dinv[dst] * H[src]
// one wave per edge: 32 lanes x float4 = coalesced 512B gather, 4 atomics/lane
// ---------------------------------------------------------------------------
__global__ __launch_bounds__(256)
void k_edge_scatter(const int* __restrict__ src, const int* __restrict__ dst,
                    const float* __restrict__ dinv, const float* __restrict__ H,
                    float* __restrict__ out, int e) {
    const int wave = threadIdx.x >> 5;
    const int lane = threadIdx.x & 31;
    const int ei = blockIdx.x * 8 + wave;
    if (ei >= e) return;
    const int s = src[ei];
    const int d = dst[ei];
    const float w = dinv[s] * dinv[d];
    const int c0 = lane * 4;
    const float4 h = *(const float4*)(H + (size_t)s * DIMS + c0);
    float* o = out + (size_t)d * DIMS + c0;
    atomicAdd(o + 0, h.x * w);
    atomicAdd(o + 1, h.y * w);
    atomicAdd(o + 2, h.z * w);
    atomicAdd(o + 3, h.w * w);
}

__global__ void k_tanh(const float* __restrict__ in, float* __restrict__ out, int total) {
    int i = blockIdx.x * blockDim.x + threadIdx.x;
    if (i < total) out[i] = tanhf(in[i]);
}

// ---------------------------------------------------------------------------
// Orchestration
// ---------------------------------------------------------------------------
extern "C" void kernel_launch(void* const* d_in, const int* in_sizes, int n_in,
                              void* d_out, int out_size, void* d_ws, size_t ws_size,
                              hipStream_t stream) {
    const float* x   = (const float*)d_in[0];
    const int*   ei  = (const int*)  d_in[1];
    const float* W1  = (const float*)d_in[2];
    const float* b1  = (const float*)d_in[3];
    const float* W2  = (const float*)d_in[4];
    const float* b2  = (const float*)d_in[5];

    const int n = in_sizes[0] / DIMS;    // 100000
    const int e = in_sizes[1] / 2;       // 500000
    const int* src = ei;
    const int* dst = ei + e;

    // workspace: deg[n] | dinv[n] | bufH[n*128] | bufA[n*128]
    float* ws   = (float*)d_ws;
    float* deg  = ws;
    float* dinv = ws + n;
    float* bufH = ws + 2 * (size_t)n;
    float* bufA = bufH + (size_t)n * DIMS;

    float* out1 = (float*)d_out;             // h1 (pre-activation)
    float* out2 = out1 + (size_t)n * DIMS;   // h2

    const int T = 256;
    // normalization
    k_deg_init <<<(n + T - 1) / T, T, 0, stream>>>(deg, n);
    k_deg_count<<<(e + T - 1) / T, T, 0, stream>>>(dst, deg, e);
    k_dinv     <<<(n + T - 1) / T, T, 0, stream>>>(deg, dinv, n);

    const int gemmBlocks = (n + 127) / 128;

    // ---- layer 1 ----
    gemm_wmma_f32 <<<gemmBlocks, T, 0, stream>>>(x, W1, bufH, n);
    k_self_bias   <<<((size_t)n * 32 + T - 1) / T, T, 0, stream>>>(bufH, dinv, b1, out1, n);
    k_edge_scatter<<<(e + 7) / 8, T, 0, stream>>>(src, dst, dinv, bufH, out1, e);

    // activation
    k_tanh<<<((size_t)n * DIMS + T - 1) / T, T, 0, stream>>>(out1, bufA, n * DIMS);

    // ---- layer 2 ----
    gemm_wmma_f32 <<<gemmBlocks, T, 0, stream>>>(bufA, W2, bufH, n);
    k_self_bias   <<<((size_t)n * 32 + T - 1) / T, T, 0, stream>>>(bufH, dinv, b2, out2, n);
    k_edge_scatter<<<(e + 7) / 8, T, 0, stream>>>(src, dst, dinv, bufH, out2, e);
}